// Wav2VecASRLoss_16054587752473
// MI455X (gfx1250) — compile-verified
//
#include <hip/hip_runtime.h>
#include <stdint.h>

// Problem constants (from the reference): B=64, T=2048, V=256, L=256
#define B_ 64
#define T_ 2048
#define V_ 256
#define L_ 256
#define S_ 513            // 2*L+1 CTC states
#define NEGF (-1e30f)

// log(exp(x)+exp(y)) with -1e30 as -inf surrogate (matches jnp.logaddexp use)
__device__ __forceinline__ float lae(float x, float y) {
    float m = fmaxf(x, y);
    float d = fminf(x, y) - m;        // <= 0 always (0 when x==y)
    return m + log1pf(__expf(d));     // exp underflows to 0 for very negative d
}

// -------- Kernel A: per-row logsumexp of logits (log_softmax denominator) --------
// One wave32 per row of 256 floats; 8 waves per block. Bandwidth-bound (128 MiB read).
__global__ __launch_bounds__(256) void lse_kernel(const float* __restrict__ logits,
                                                  float* __restrict__ lse) {
    const int wave = threadIdx.x >> 5;
    const int lane = threadIdx.x & 31;
    const int row  = blockIdx.x * 8 + wave;          // row in [0, B*T)
    const float* p = logits + (size_t)row * V_;
    // lane covers cols [lane*8, lane*8+8): two float4 loads, fully coalesced per wave
    float4 a = ((const float4*)p)[lane * 2 + 0];
    float4 b = ((const float4*)p)[lane * 2 + 1];
    float mx = fmaxf(fmaxf(fmaxf(a.x, a.y), fmaxf(a.z, a.w)),
                     fmaxf(fmaxf(b.x, b.y), fmaxf(b.z, b.w)));
    #pragma unroll
    for (int off = 16; off > 0; off >>= 1)
        mx = fmaxf(mx, __shfl_xor(mx, off, 32));
    float s = __expf(a.x - mx) + __expf(a.y - mx) + __expf(a.z - mx) + __expf(a.w - mx)
            + __expf(b.x - mx) + __expf(b.y - mx) + __expf(b.z - mx) + __expf(b.w - mx);
    #pragma unroll
    for (int off = 16; off > 0; off >>= 1)
        s += __shfl_xor(s, off, 32);
    if (lane == 0) lse[row] = mx + logf(s);
}

// -------- Kernel B: CTC alpha recursion, one workgroup per batch element --------
// 544 threads = 17 waves; thread s owns state s (s < 513). Alpha double-buffered in
// LDS; logits row for step t+1 async-copied global->LDS (triple-buffered) while
// step t computes; lse[b,:] fully staged in LDS at setup.
__global__ __launch_bounds__(544) void ctc_alpha_kernel(
    const float* __restrict__ logits,
    const int*   __restrict__ labels,
    const int*   __restrict__ input_lengths,
    const float* __restrict__ lse,
    float*       __restrict__ per_batch) {

    __shared__ float         s_lse[T_];            // 8 KB
    __shared__ float         rowbuf[3][V_];        // 3 KB, async landing buffers
    __shared__ float         A[2][S_ + 7];         // alpha, state s stored at [s+2]
    __shared__ short         ext[S_];              // extended label sequence
    __shared__ unsigned char skip[S_];             // allow_skip mask
    __shared__ int           s_tlen;

    const int b    = blockIdx.x;
    const int tid  = threadIdx.x;
    const int ilen = input_lengths[b];

    if (tid == 0) s_tlen = 0;
    __syncthreads();

    // target length = count(labels >= 0); LDS atomic (integer -> deterministic)
    if (tid < L_) {
        if (labels[b * L_ + tid] >= 0) atomicAdd(&s_tlen, 1);
    }
    // stage lse row for this batch
    for (int i = tid; i < T_; i += 544) s_lse[i] = lse[(size_t)b * T_ + i];
    __syncthreads();
    const int tlen  = s_tlen;
    const int smax  = 2 * tlen + 1;                // states < smax are valid

    // extended labels: blank, l1, blank, l2, ..., blank
    if (tid < S_) {
        int e = 0;
        if (tid & 1) {
            int lab = labels[b * L_ + ((tid - 1) >> 1)];
            e = lab >= 0 ? lab : 0;
        }
        ext[tid] = (short)e;
    }
    __syncthreads();
    if (tid < S_) {
        int e = ext[tid];
        skip[tid] = (unsigned char)((tid >= 2) && (e != 0) && (e != ext[tid - 2]));
    }
    if (tid < 2) { A[0][tid] = NEGF; A[1][tid] = NEGF; }   // alpha[-1], alpha[-2] pads

    // alpha0 from row 0 (direct global gather; one-time)
    if (tid < S_) {
        float e0 = logits[((size_t)b * T_) * V_ + (int)ext[tid]] - s_lse[0];
        A[0][tid + 2] = (tid < 2 && tid < smax) ? e0 : NEGF;
    }

    // Async-prefetch row 1 -> rowbuf[1]: 64 lanes x 16B = 1 KB (waves 0,1 only)
    if (tid < 64) {
        unsigned lo = (unsigned)(uintptr_t)(&rowbuf[1][0]) + (unsigned)tid * 16u;
        unsigned long long ga =
            (unsigned long long)(uintptr_t)(logits + ((size_t)b * T_ + 1) * V_) + (unsigned long long)tid * 16u;
        asm volatile("global_load_async_to_lds_b128 %0, %1, off"
                     :: "v"(lo), "v"(ga) : "memory");
    }
    __syncthreads();

    int cur = 0;
    for (int t = 1; t < T_; ++t) {
        const int nb = t % 3;                       // buffer holding row t
        if (tid < 64) {                             // whole waves 0,1 -> no divergence
            if (t + 1 < T_) {
                unsigned lo = (unsigned)(uintptr_t)(&rowbuf[(t + 1) % 3][0]) + (unsigned)tid * 16u;
                unsigned long long ga =
                    (unsigned long long)(uintptr_t)(logits + ((size_t)b * T_ + (t + 1)) * V_) + (unsigned long long)tid * 16u;
                asm volatile("global_load_async_to_lds_b128 %0, %1, off"
                             :: "v"(lo), "v"(ga) : "memory");
                // async ops retire in order: cnt<=1 => row t landed, row t+1 in flight
                asm volatile("s_wait_asynccnt 1" ::: "memory");
            } else {
                asm volatile("s_wait_asynccnt 0" ::: "memory");
            }
        }
        __syncthreads();  // row t visible to all; prev alpha writes visible

        if (tid < S_) {
            float e  = rowbuf[nb][(int)ext[tid]] - s_lse[t];
            float a  = A[cur][tid + 2];
            float a1 = A[cur][tid + 1];
            float a2 = skip[tid] ? A[cur][tid] : NEGF;
            float nv = lae(lae(a, a1), a2) + e;
            if (tid >= smax) nv = NEGF;
            A[cur ^ 1][tid + 2] = (t < ilen) ? nv : a;
        }
        cur ^= 1;
    }
    __syncthreads();

    if (tid == 0) {
        float e1 = A[cur][2 * tlen + 2];            // last blank
        float e2 = A[cur][2 * tlen + 1];            // last label
        float ll = lae(e1, e2);
        per_batch[b] = (ll > NEGF * 0.5f) ? (-ll / (float)tlen) : 0.0f;
    }
}

// -------- Kernel C: deterministic mean over batch --------
__global__ void finalize_kernel(const float* __restrict__ per_batch,
                                float* __restrict__ out) {
    if (threadIdx.x == 0) {
        float s = 0.0f;
        for (int i = 0; i < B_; ++i) s += per_batch[i];
        out[0] = s / (float)B_;
    }
}

extern "C" void kernel_launch(void* const* d_in, const int* in_sizes, int n_in,
                              void* d_out, int out_size, void* d_ws, size_t ws_size,
                              hipStream_t stream) {
    const float* logits        = (const float*)d_in[0];   // [B,T,V] f32
    const int*   labels        = (const int*)d_in[1];     // [B,L]   i32
    const int*   input_lengths = (const int*)d_in[2];     // [B]     i32

    float* lse       = (float*)d_ws;                      // B*T floats (512 KB)
    float* per_batch = lse + (size_t)B_ * T_;             // B floats

    lse_kernel<<<(B_ * T_) / 8, 256, 0, stream>>>(logits, lse);
    ctc_alpha_kernel<<<B_, 544, 0, stream>>>(logits, labels, input_lengths, lse, per_batch);
    finalize_kernel<<<1, 64, 0, stream>>>(per_batch, (float*)d_out);
}